// OSPNet_56616258896179
// MI455X (gfx1250) — compile-verified
//
#include <hip/hip_runtime.h>
#include <hip/hip_bf16.h>

// ---------------------------------------------------------------------------
// OSPNet on MI455X (gfx1250, wave32).
// eigh -> Newton-Schulz inverse-scaling-and-squaring matrix log (pure GEMM),
// every GEMM through V_WMMA_F32_16X16X4_F32. The dominant row-major 256^3
// batched GEMMs use GLOBAL_LOAD_ASYNC_TO_LDS_B128 double-buffered panels
// (ASYNCcnt pipeline); strided variants use the LDS-staged generic kernel.
// ---------------------------------------------------------------------------

typedef float v2f __attribute__((ext_vector_type(2)));
typedef float v8f __attribute__((ext_vector_type(8)));

#define BB     32          // batch
#define CIN    2048
#define HW     34
#define POOLN  17
#define NPOS   289         // 17*17
#define NPAD   304         // 19*16, zero-padded rows
#define CR     256
#define FEAT   32896       // 256*257/2
#define KLEV   5           // number of NS square-root levels
#define NSIT   10          // Newton-Schulz iterations per sqrt
#define LOGT   8           // Gregory series terms for log near I

__device__ __forceinline__ int imin(int a, int b) { return a < b ? a : b; }

// ---------------------------------------------------------------------------
// 2x2 average pool: x[B,Cin,34,34] -> P[B,Cin,289]  (n = ph*17+pw fastest)
// ---------------------------------------------------------------------------
__global__ void pool2x2_kernel(const float* __restrict__ x,
                               float* __restrict__ P, int total) {
    int i = blockIdx.x * blockDim.x + threadIdx.x;
    if (i >= total) return;
    int n  = i % NPOS;
    long long bc = i / NPOS;
    int pw = n % POOLN, ph = n / POOLN;
    const float* s = x + (bc * HW + 2 * ph) * HW + 2 * pw;
    P[i] = 0.25f * (s[0] + s[1] + s[HW] + s[HW + 1]);
}

// ---------------------------------------------------------------------------
// Specialized row-major x row-major batched GEMM with CDNA5 async-to-LDS.
//   C[b] = alpha * A[b] (M x K, lda) * B[b] (K x N, ldb)
// Requires: M,N multiples of 64; K multiple of 16; lda,ldb multiples of 4.
//
// Block = 128 threads (4 waves), 64x64 C tile, K chunked by 16,
// double-buffered LDS filled by GLOBAL_LOAD_ASYNC_TO_LDS_B128 (ASYNCcnt).
// Pipeline: issue chunk i+1's 4 async ops, then s_wait_asynccnt 4 retires
// chunk i (async loads complete in order), barrier, 16 WMMA, barrier.
//
// LDS layouts:
//   lA[buf][row][20]  row pad 20 floats = 80 B: 16B-aligned b128 dests, and
//                     stride-20 banks are distinct across 16 lanes for the
//                     b64 fragment loads (k even -> 8B aligned).
//   lB[buf][k][64]    256 B rows: aligned b128 dests; fragment = two b32 at
//                     +0/+256 B (lane-contiguous, conflict-free).
// ---------------------------------------------------------------------------
__global__ __launch_bounds__(128)
void wmma_gemm_nn_async(const float* __restrict__ A, const float* __restrict__ B,
                        float* __restrict__ C, int M, int N, int K,
                        long long lda, long long ldb, long long ldc,
                        long long batchA, long long batchB, long long batchC,
                        float alpha) {
    __shared__ __align__(16) float lA[2][64 * 20];
    __shared__ __align__(16) float lB[2][16 * 64];

    const int tid  = threadIdx.x;
    const int lane = tid & 31;
    const int wave = tid >> 5;
    const int sub  = lane & 15;
    const int half = lane >> 4;
    const int m0 = blockIdx.y * 64;
    const int n0 = blockIdx.x * 64;
    const int wm = (wave >> 1) * 32;
    const int wn = (wave & 1) * 32;

    const float* Ab = A + (long long)blockIdx.z * batchA;
    const float* Bb = B + (long long)blockIdx.z * batchB;
    float*       Cb = C + (long long)blockIdx.z * batchC;

    const int rA  = tid >> 2;   // A panel: row (and row+32), kq = tid&3
    const int kqA = tid & 3;
    const int kB  = tid >> 4;   // B panel: k (and k+8), nq = tid&15
    const int nqB = tid & 15;

    v8f acc00 = {}, acc01 = {}, acc10 = {}, acc11 = {};

#define ISSUE_CHUNK(kb, buf)                                                   \
    do {                                                                       \
        const float* ga0 = &Ab[(long long)(m0 + rA) * lda + (kb) + 4 * kqA];   \
        const float* ga1 = &Ab[(long long)(m0 + rA + 32) * lda + (kb) + 4 * kqA]; \
        unsigned da0 = (unsigned)(unsigned long long)&lA[buf][rA * 20 + 4 * kqA]; \
        unsigned da1 = (unsigned)(unsigned long long)&lA[buf][(rA + 32) * 20 + 4 * kqA]; \
        const float* gb0 = &Bb[(long long)((kb) + kB) * ldb + n0 + 4 * nqB];   \
        const float* gb1 = &Bb[(long long)((kb) + kB + 8) * ldb + n0 + 4 * nqB]; \
        unsigned db0 = (unsigned)(unsigned long long)&lB[buf][kB * 64 + 4 * nqB]; \
        unsigned db1 = (unsigned)(unsigned long long)&lB[buf][(kB + 8) * 64 + 4 * nqB]; \
        asm volatile("global_load_async_to_lds_b128 %0, %1, off"               \
                     :: "v"(da0), "v"(ga0) : "memory");                        \
        asm volatile("global_load_async_to_lds_b128 %0, %1, off"               \
                     :: "v"(da1), "v"(ga1) : "memory");                        \
        asm volatile("global_load_async_to_lds_b128 %0, %1, off"               \
                     :: "v"(db0), "v"(gb0) : "memory");                        \
        asm volatile("global_load_async_to_lds_b128 %0, %1, off"               \
                     :: "v"(db1), "v"(gb1) : "memory");                        \
    } while (0)

    ISSUE_CHUNK(0, 0);
    const int nchunk = K / 16;
    for (int i = 0; i < nchunk; ++i) {
        const int buf = i & 1;
        if (i + 1 < nchunk) {
            ISSUE_CHUNK((i + 1) * 16, buf ^ 1);
            asm volatile("s_wait_asynccnt 0x4" ::: "memory");  // chunk i done
        } else {
            asm volatile("s_wait_asynccnt 0x0" ::: "memory");
        }
        __syncthreads();   // all waves' chunk-i LDS writes visible
#pragma unroll
        for (int s = 0; s < 4; ++s) {
            const int k = s * 4 + half * 2;    // lane-half selects K pair
            v2f a0 = *(const v2f*)&lA[buf][(wm + sub) * 20 + k];
            v2f a1 = *(const v2f*)&lA[buf][(wm + 16 + sub) * 20 + k];
            v2f b0 = { lB[buf][k * 64 + wn + sub],
                       lB[buf][(k + 1) * 64 + wn + sub] };
            v2f b1 = { lB[buf][k * 64 + wn + 16 + sub],
                       lB[buf][(k + 1) * 64 + wn + 16 + sub] };
            acc00 = __builtin_amdgcn_wmma_f32_16x16x4_f32(false, a0, false, b0,
                                                          (short)0, acc00, false, false);
            acc01 = __builtin_amdgcn_wmma_f32_16x16x4_f32(false, a0, false, b1,
                                                          (short)0, acc01, false, false);
            acc10 = __builtin_amdgcn_wmma_f32_16x16x4_f32(false, a1, false, b0,
                                                          (short)0, acc10, false, false);
            acc11 = __builtin_amdgcn_wmma_f32_16x16x4_f32(false, a1, false, b1,
                                                          (short)0, acc11, false, false);
        }
        __syncthreads();   // compute done before this buf is refilled
    }
#undef ISSUE_CHUNK

    const int colb = n0 + wn + sub;
#pragma unroll
    for (int mi = 0; mi < 2; ++mi) {
        int row0 = m0 + wm + mi * 16 + half * 8;
#pragma unroll
        for (int ni = 0; ni < 2; ++ni) {
            const v8f* a = (mi == 0) ? (ni == 0 ? &acc00 : &acc01)
                                     : (ni == 0 ? &acc10 : &acc11);
            int col = colb + ni * 16;
#pragma unroll
            for (int v = 0; v < 8; ++v)
                Cb[(long long)(row0 + v) * ldc + col] = alpha * (*a)[v];
        }
    }
}

// ---------------------------------------------------------------------------
// Generic batched strided GEMM, fp32 WMMA 16x16x4, LDS-staged (for the
// strided/transposed cases: channel reduce, covariance F^T F, MLP).
//   C[b][m*ldc+n] = alpha * sum_k A[b][m*rsA+k*csA] * B[b][k*rsB+n*csB]
// AMODE 0: csA==1 (float4 along k)   AMODE 1: rsA==1 (m contiguous)
// BMODE 0: csB==1 (float4 along n)   BMODE 1: rsB==1 (float4 along k)
// ---------------------------------------------------------------------------
template <int AMODE, int BMODE>
__global__ __launch_bounds__(128)
void wmma_gemm_f32(const float* __restrict__ A, const float* __restrict__ B,
                   float* __restrict__ C,
                   int M, int N, int K,
                   long long rsA, long long csA,
                   long long rsB, long long csB, long long ldc,
                   long long batchA, long long batchB, long long batchC,
                   float alpha) {
    __shared__ float lA[8 * 64 * 2];   // [kpair][m][2]
    __shared__ float lB[8 * 64 * 2];   // [kpair][n][2]

    const int tid  = threadIdx.x;
    const int lane = tid & 31;
    const int wave = tid >> 5;
    const int sub  = lane & 15;
    const int half = lane >> 4;

    const int m0 = blockIdx.y * 64;
    const int n0 = blockIdx.x * 64;
    const int wm = (wave >> 1) * 32;
    const int wn = (wave & 1) * 32;

    const float* Ab = A + (long long)blockIdx.z * batchA;
    const float* Bb = B + (long long)blockIdx.z * batchB;
    float*       Cb = C + (long long)blockIdx.z * batchC;

    v8f acc00 = {}, acc01 = {}, acc10 = {}, acc11 = {};

    for (int kb = 0; kb < K; kb += 16) {
        float4 ra[2], rb[2];
#pragma unroll
        for (int it = 0; it < 2; ++it) {
            if (AMODE == 0) {
                int q = tid + it * 128, r = q >> 2, kq = q & 3;
                int mr = imin(m0 + r, M - 1);
                ra[it] = *(const float4*)&Ab[(long long)mr * rsA + (kb + kq * 4)];
            } else {
                int g = tid + it * 128, k = g >> 4, mq = g & 15;
                long long kk = (long long)(kb + k) * csA;
                int mA = m0 + mq * 4;
                ra[it].x = Ab[imin(mA + 0, M - 1) + kk];
                ra[it].y = Ab[imin(mA + 1, M - 1) + kk];
                ra[it].z = Ab[imin(mA + 2, M - 1) + kk];
                ra[it].w = Ab[imin(mA + 3, M - 1) + kk];
            }
            if (BMODE == 0) {
                int g = tid + it * 128, k = g >> 4, nq = g & 15;
                rb[it] = *(const float4*)&Bb[(long long)(kb + k) * rsB + (n0 + nq * 4)];
            } else {
                int q = tid + it * 128, n = q >> 2, kq = q & 3;
                rb[it] = *(const float4*)&Bb[(long long)(n0 + n) * csB + (kb + kq * 4)];
            }
        }
        __syncthreads();
#pragma unroll
        for (int it = 0; it < 2; ++it) {
            if (AMODE == 0) {
                int q = tid + it * 128, r = q >> 2, kq = q & 3;
                *(v2f*)&lA[((2 * kq + 0) * 64 + r) * 2] = (v2f){ra[it].x, ra[it].y};
                *(v2f*)&lA[((2 * kq + 1) * 64 + r) * 2] = (v2f){ra[it].z, ra[it].w};
            } else {
                int g = tid + it * 128, k = g >> 4, mq = g & 15;
                int base = ((k >> 1) * 64 + mq * 4) * 2 + (k & 1);
                lA[base + 0] = ra[it].x;
                lA[base + 2] = ra[it].y;
                lA[base + 4] = ra[it].z;
                lA[base + 6] = ra[it].w;
            }
            if (BMODE == 0) {
                int g = tid + it * 128, k = g >> 4, nq = g & 15;
                int base = ((k >> 1) * 64 + nq * 4) * 2 + (k & 1);
                lB[base + 0] = rb[it].x;
                lB[base + 2] = rb[it].y;
                lB[base + 4] = rb[it].z;
                lB[base + 6] = rb[it].w;
            } else {
                int q = tid + it * 128, n = q >> 2, kq = q & 3;
                *(v2f*)&lB[((2 * kq + 0) * 64 + n) * 2] = (v2f){rb[it].x, rb[it].y};
                *(v2f*)&lB[((2 * kq + 1) * 64 + n) * 2] = (v2f){rb[it].z, rb[it].w};
            }
        }
        __syncthreads();
#pragma unroll
        for (int s = 0; s < 4; ++s) {
            const int pp = s * 2 + half;
            v2f a0 = *(const v2f*)&lA[(pp * 64 + wm + 0  + sub) * 2];
            v2f a1 = *(const v2f*)&lA[(pp * 64 + wm + 16 + sub) * 2];
            v2f b0 = *(const v2f*)&lB[(pp * 64 + wn + 0  + sub) * 2];
            v2f b1 = *(const v2f*)&lB[(pp * 64 + wn + 16 + sub) * 2];
            acc00 = __builtin_amdgcn_wmma_f32_16x16x4_f32(false, a0, false, b0,
                                                          (short)0, acc00, false, false);
            acc01 = __builtin_amdgcn_wmma_f32_16x16x4_f32(false, a0, false, b1,
                                                          (short)0, acc01, false, false);
            acc10 = __builtin_amdgcn_wmma_f32_16x16x4_f32(false, a1, false, b0,
                                                          (short)0, acc10, false, false);
            acc11 = __builtin_amdgcn_wmma_f32_16x16x4_f32(false, a1, false, b1,
                                                          (short)0, acc11, false, false);
        }
        __syncthreads();
    }

    const int colb = n0 + wn + sub;
#pragma unroll
    for (int mi = 0; mi < 2; ++mi) {
        int row0 = m0 + wm + mi * 16 + half * 8;
#pragma unroll
        for (int ni = 0; ni < 2; ++ni) {
            const v8f* a = (mi == 0) ? (ni == 0 ? &acc00 : &acc01)
                                     : (ni == 0 ? &acc10 : &acc11);
            int col = colb + ni * 16;
#pragma unroll
            for (int v = 0; v < 8; ++v) {
                int row = row0 + v;
                if (row < M)
                    Cb[(long long)row * ldc + col] = alpha * (*a)[v];
            }
        }
    }
}

// ---------------------------------------------------------------------------
// Small kernels
// ---------------------------------------------------------------------------
__global__ void zero_rows_kernel(float* __restrict__ F) {
    int b = blockIdx.x, r = NPOS + blockIdx.y, d = threadIdx.x;
    F[(long long)b * NPAD * CR + (long long)r * CR + d] = 0.0f;
}

__global__ void center_kernel(float* __restrict__ F) {
    int b = blockIdx.x, d = threadIdx.x;
    float* Fb = F + (long long)b * NPAD * CR;
    float s = 0.0f;
    for (int n = 0; n < NPOS; ++n) s += Fb[n * CR + d];
    float mean = s * (1.0f / (float)NPOS);
    for (int n = 0; n < NPOS; ++n) Fb[n * CR + d] -= mean;
}

__global__ void trace_log_kernel(const float* __restrict__ covA,
                                 float* __restrict__ S) {
    int b = blockIdx.x, t = threadIdx.x;
    float v = covA[(long long)b * CR * CR + (long long)t * (CR + 1)];
    __shared__ float red[8];
#pragma unroll
    for (int o = 16; o > 0; o >>= 1) v += __shfl_down(v, o, 32);
    if ((t & 31) == 0) red[t >> 5] = v;
    __syncthreads();
    if (t == 0) {
        float s = 0.0f;
#pragma unroll
        for (int i = 0; i < 8; ++i) s += red[i];
        S[b]      = logf(s);
        S[32 + b] = s;
    }
}

__global__ void ew_axpyi_kernel(const float* __restrict__ in,
                                float* __restrict__ out,
                                const float* __restrict__ S,
                                float a, float c0, float c1, int divTrace) {
    int b = blockIdx.x;
    int idx = blockIdx.y * blockDim.x + threadIdx.x;   // 0..65535
    int i = idx >> 8, j = idx & 255;
    float scale = divTrace ? (a / S[32 + b]) : a;
    long long o = (long long)b * CR * CR + idx;
    float v = scale * in[o];
    if (i == j) v += c0 + c1 * S[b];
    out[o] = v;
}

__global__ void flatten_triu_kernel(const float* __restrict__ L,
                                    float* __restrict__ flat) {
    int b = blockIdx.x, i = blockIdx.y, j = threadIdx.x;
    if (j < i) return;
    long long base = (long long)i * CR - (long long)i * (i - 1) / 2;
    float v = L[(long long)b * CR * CR + (long long)i * CR + j];
    flat[(long long)b * FEAT + base + (j - i)] =
        (i == j) ? v : v * 1.41421356237309515f;
}

__global__ void bias_relu_kernel(float* __restrict__ h,
                                 const float* __restrict__ b1) {
    int b = blockIdx.x, o = threadIdx.x;
    float v = h[b * CR + o] + b1[o];
    h[b * CR + o] = v > 0.0f ? v : 0.0f;
}

__global__ void mlp2_kernel(const float* __restrict__ h,
                            const float* __restrict__ W2,
                            const float* __restrict__ b2,
                            float* __restrict__ out) {
    int b = blockIdx.x, c = blockIdx.y, t = threadIdx.x;
    float v = h[b * CR + t] * W2[c * CR + t];
    __shared__ float red[8];
#pragma unroll
    for (int o = 16; o > 0; o >>= 1) v += __shfl_down(v, o, 32);
    if ((t & 31) == 0) red[t >> 5] = v;
    __syncthreads();
    if (t == 0) {
        float s = 0.0f;
#pragma unroll
        for (int i = 0; i < 8; ++i) s += red[i];
        out[b * 3 + c] = s + b2[c];
    }
}

// ---------------------------------------------------------------------------
// Host-side orchestration
// ---------------------------------------------------------------------------
static inline void launch_gemm(hipStream_t st, int amode, int bmode,
                               const float* A, const float* B, float* C,
                               int M, int N, int K,
                               long long rsA, long long csA,
                               long long rsB, long long csB, long long ldc,
                               long long bA, long long bB, long long bC,
                               int nb, float alpha) {
    dim3 grid(N / 64, (M + 63) / 64, nb);
    dim3 blk(128);
    if (amode == 0 && bmode == 0)   // row-major x row-major -> async pipeline
        hipLaunchKernelGGL(wmma_gemm_nn_async, grid, blk, 0, st, A, B, C,
                           M, N, K, rsA, rsB, ldc, bA, bB, bC, alpha);
    else if (amode == 0 && bmode == 1)
        hipLaunchKernelGGL((wmma_gemm_f32<0, 1>), grid, blk, 0, st, A, B, C,
                           M, N, K, rsA, csA, rsB, csB, ldc, bA, bB, bC, alpha);
    else if (amode == 1 && bmode == 0)
        hipLaunchKernelGGL((wmma_gemm_f32<1, 0>), grid, blk, 0, st, A, B, C,
                           M, N, K, rsA, csA, rsB, csB, ldc, bA, bB, bC, alpha);
    else
        hipLaunchKernelGGL((wmma_gemm_f32<1, 1>), grid, blk, 0, st, A, B, C,
                           M, N, K, rsA, csA, rsB, csB, ldc, bA, bB, bC, alpha);
}

extern "C" void kernel_launch(void* const* d_in, const int* in_sizes, int n_in,
                              void* d_out, int out_size, void* d_ws, size_t ws_size,
                              hipStream_t stream) {
    const float* x     = (const float*)d_in[0];
    const float* W_red = (const float*)d_in[1];
    const float* W1    = (const float*)d_in[2];
    const float* b1    = (const float*)d_in[3];
    const float* W2    = (const float*)d_in[4];
    const float* b2    = (const float*)d_in[5];
    float* out = (float*)d_out;

    // workspace carve-up (floats)
    float* P  = (float*)d_ws;                              // [32,2048,289]
    float* F  = P  + (long long)BB * CIN * NPOS;           // [32,304,256]
    float* M0 = F  + (long long)BB * NPAD * CR;            // [32,256,256]
    float* Y  = M0 + (long long)BB * CR * CR;
    float* Z  = Y  + (long long)BB * CR * CR;
    float* T  = Z  + (long long)BB * CR * CR;
    float* Y2 = T  + (long long)BB * CR * CR;
    float* Z2 = Y2 + (long long)BB * CR * CR;
    float* fl = Z2 + (long long)BB * CR * CR;              // [32,32896]
    float* h0 = fl + (long long)BB * FEAT;                 // [32,256]
    float* S  = h0 + (long long)BB * CR;                   // [64]

    const long long MM = (long long)CR * CR;               // 65536

    // 1) pool (x: 303 MB @ 23.3 TB/s ~ 13 us; P (76 MB) then lives in L2)
    {
        int total = BB * CIN * NPOS;
        hipLaunchKernelGGL(pool2x2_kernel, dim3((total + 255) / 256), dim3(256),
                           0, stream, x, P, total);
    }

    // 2) channel reduce: F[b][n][d] = sum_c P[b][c][n] * W_red[d][c]
    launch_gemm(stream, 1, 1, P, W_red, F, NPAD, CR, CIN,
                1, NPOS, 1, CIN, CR,
                (long long)CIN * NPOS, 0, (long long)NPAD * CR, BB, 1.0f);

    // 3) zero pad rows, then center columns over the 289 valid rows
    hipLaunchKernelGGL(zero_rows_kernel, dim3(BB, NPAD - NPOS), dim3(CR),
                       0, stream, F);
    hipLaunchKernelGGL(center_kernel, dim3(BB), dim3(CR), 0, stream, F);

    // 4) covariance: cov = F^T F / (N-1)   (pad rows exact zero)
    launch_gemm(stream, 1, 0, F, F, M0, CR, CR, NPAD,
                1, CR, CR, 1, CR,
                (long long)NPAD * CR, (long long)NPAD * CR, MM, BB,
                1.0f / (float)(NPOS - 1));

    // 5) trace + log(trace)
    hipLaunchKernelGGL(trace_log_kernel, dim3(BB), dim3(CR), 0, stream, M0, S);

    dim3 ewg(BB, MM / 256);
    // A0 = cov / trace  (spectrum in (0,1])
    hipLaunchKernelGGL(ew_axpyi_kernel, ewg, dim3(256), 0, stream,
                       M0, M0, S, 1.0f, 0.0f, 0.0f, 1);

    // 6) KLEV Newton-Schulz square roots: A <- sqrt(A)  (async NN kernel)
    for (int lev = 0; lev < KLEV; ++lev) {
        hipLaunchKernelGGL(ew_axpyi_kernel, ewg, dim3(256), 0, stream,
                           M0, Y, S, 1.0f, 0.0f, 0.0f, 0);   // Y = A
        hipLaunchKernelGGL(ew_axpyi_kernel, ewg, dim3(256), 0, stream,
                           M0, Z, S, 0.0f, 1.0f, 0.0f, 0);   // Z = I
        float* y = Y; float* z = Z; float* y2 = Y2; float* z2 = Z2;
        for (int it = 0; it < NSIT; ++it) {
            launch_gemm(stream, 0, 0, z, y, T, CR, CR, CR,
                        CR, 1, CR, 1, CR, MM, MM, MM, BB, 1.0f);   // T = Z*Y
            hipLaunchKernelGGL(ew_axpyi_kernel, ewg, dim3(256), 0, stream,
                               T, T, S, -0.5f, 1.5f, 0.0f, 0);     // T = 1.5I-0.5T
            launch_gemm(stream, 0, 0, y, T, y2, CR, CR, CR,
                        CR, 1, CR, 1, CR, MM, MM, MM, BB, 1.0f);   // Y' = Y*T
            launch_gemm(stream, 0, 0, T, z, z2, CR, CR, CR,
                        CR, 1, CR, 1, CR, MM, MM, MM, BB, 1.0f);   // Z' = T*Z
            float* t0 = y; y = y2; y2 = t0;
            float* t1 = z; z = z2; z2 = t1;
        }
        hipLaunchKernelGGL(ew_axpyi_kernel, ewg, dim3(256), 0, stream,
                           y, M0, S, 1.0f, 0.0f, 0.0f, 0);         // A = sqrt(A)
    }

    // 7) log near identity: E = R - I, Gregory series via Horner
    float* E = Z;
    hipLaunchKernelGGL(ew_axpyi_kernel, ewg, dim3(256), 0, stream,
                       M0, E, S, 1.0f, -1.0f, 0.0f, 0);
    float coef[LOGT + 1];
    for (int i = 1; i <= LOGT; ++i) coef[i] = ((i & 1) ? 1.0f : -1.0f) / (float)i;
    float* p = Y2; float* q = Z2;
    hipLaunchKernelGGL(ew_axpyi_kernel, ewg, dim3(256), 0, stream,
                       M0, p, S, 0.0f, coef[LOGT], 0.0f, 0);       // p = c_T I
    for (int t = LOGT - 1; t >= 1; --t) {
        launch_gemm(stream, 0, 0, E, p, q, CR, CR, CR,
                    CR, 1, CR, 1, CR, MM, MM, MM, BB, 1.0f);       // q = E*p
        hipLaunchKernelGGL(ew_axpyi_kernel, ewg, dim3(256), 0, stream,
                           q, q, S, 1.0f, coef[t], 0.0f, 0);       // p = q + c_t I
        float* tt = p; p = q; q = tt;
    }
    launch_gemm(stream, 0, 0, E, p, T, CR, CR, CR,
                CR, 1, CR, 1, CR, MM, MM, MM, BB, 1.0f);           // logR = E*p
    // logm(cov) = 2^KLEV * logR + log(trace) * I
    hipLaunchKernelGGL(ew_axpyi_kernel, ewg, dim3(256), 0, stream,
                       T, M0, S, (float)(1 << KLEV), 0.0f, 1.0f, 0);

    // 8) triu flatten with sqrt(2)
    hipLaunchKernelGGL(flatten_triu_kernel, dim3(BB, CR), dim3(CR), 0, stream,
                       M0, fl);

    // 9) MLP: h = relu(flat @ W1^T + b1); out = h @ W2^T + b2
    launch_gemm(stream, 0, 1, fl, W1, h0, BB, CR, FEAT,
                FEAT, 1, 1, FEAT, CR, 0, 0, 0, 1, 1.0f);
    hipLaunchKernelGGL(bias_relu_kernel, dim3(BB), dim3(CR), 0, stream, h0, b1);
    hipLaunchKernelGGL(mlp2_kernel, dim3(BB, 3), dim3(256), 0, stream,
                       h0, W2, b2, out);
}